// MarketGAT_32581621907636
// MI455X (gfx1250) — compile-verified
//
#include <hip/hip_runtime.h>

typedef float v2f __attribute__((ext_vector_type(2)));
typedef float v8f __attribute__((ext_vector_type(8)));

#define NEG_SLOPE 0.2f

__device__ __forceinline__ float lrelu(float x) { return x > 0.f ? x : NEG_SLOPE * x; }

// Float atomic max via sign-split integer atomics (monotone bit encoding).
__device__ __forceinline__ void atomicMaxF(float* addr, float value) {
  if (value >= 0.f) atomicMax((int*)addr, __float_as_int(value));
  else              atomicMin((unsigned int*)addr, __float_as_uint(value));
}

// ---------------------------------------------------------------------------
// C[M,N] = A[M,K] * B[K,N], fp32 WMMA 16x16x4.
// Each wave computes a 16 x (16*CT) strip: one A-fragment load feeds CT WMMAs
// per K-step (A reuse, higher matrix-op density per loop iteration).
// Requires M%16==0, N%(16*CT)==0, K%4==0.
// A-frag: lane<16 holds (A[m][k],A[m][k+1]); lane>=16 holds (A[m][k+2],A[m][k+3]).
// B-frag mirrors with N across lanes. C: VGPR v -> row m0+v+8*(lane>=16).
// ---------------------------------------------------------------------------
template <int CT>
__global__ void gemm_f32_wmma(const float* __restrict__ A,
                              const float* __restrict__ B,
                              float* __restrict__ C,
                              int M, int K, int N) {
  const int lane = threadIdx.x & 31;
  const int wave = (blockIdx.x * blockDim.x + threadIdx.x) >> 5;
  const int groupsN = N / (16 * CT);
  const int tiles   = (M >> 4) * groupsN;
  if (wave >= tiles) return;
  const int m0  = (wave / groupsN) << 4;
  const int n0  = (wave % groupsN) * (16 * CT);
  const int l15 = lane & 15;
  const int kh  = (lane >> 4) << 1;  // 0 or 2

  const float* arow = A + (size_t)(m0 + l15) * K + kh;
  const float* bcol = B + (size_t)kh * N + n0 + l15;

  v8f acc[CT];
#pragma unroll
  for (int t = 0; t < CT; ++t) acc[t] = (v8f){0.f, 0.f, 0.f, 0.f, 0.f, 0.f, 0.f, 0.f};

  for (int k = 0; k < K; k += 4) {
    v2f a = *(const v2f*)(arow + k);
    v2f b[CT];
#pragma unroll
    for (int t = 0; t < CT; ++t) {
      b[t].x = bcol[(size_t)k * N + t * 16];
      b[t].y = bcol[(size_t)(k + 1) * N + t * 16];
    }
#pragma unroll
    for (int t = 0; t < CT; ++t) {
      acc[t] = __builtin_amdgcn_wmma_f32_16x16x4_f32(false, a, false, b[t],
                                                     (short)0, acc[t], false, false);
    }
  }

  const int crow = m0 + ((lane >> 4) << 3);
  float* cp = C + (size_t)crow * N + n0 + l15;
#pragma unroll
  for (int t = 0; t < CT; ++t)
#pragma unroll
    for (int v = 0; v < 8; ++v) cp[(size_t)v * N + t * 16] = acc[t][v];
}

// ---------------------------------------------------------------------------
// Layer 1 attention prep: warp per node, H=4 heads x 64 channels.
// Computes a_src, a_dst; initializes segment-max with the self-loop logit.
// ---------------------------------------------------------------------------
__global__ void prep1(const float* __restrict__ h1,
                      const float* __restrict__ att_src,
                      const float* __restrict__ att_dst,
                      float* __restrict__ asrc, float* __restrict__ adst,
                      float* __restrict__ eself, float* __restrict__ emax,
                      int N) {
  const int wave = (blockIdx.x * blockDim.x + threadIdx.x) >> 5;
  const int lane = threadIdx.x & 31;
  if (wave >= N) return;
  const int head = lane >> 3;
  const float4* hv = (const float4*)(h1 + (size_t)wave * 256 + lane * 8);
  const float4* sv = (const float4*)(att_src + lane * 8);
  const float4* dv = (const float4*)(att_dst + lane * 8);
  float4 h0 = hv[0], h1v = hv[1];
  float4 s0 = sv[0], s1v = sv[1];
  float4 d0 = dv[0], d1v = dv[1];
  float ps = h0.x*s0.x + h0.y*s0.y + h0.z*s0.z + h0.w*s0.w
           + h1v.x*s1v.x + h1v.y*s1v.y + h1v.z*s1v.z + h1v.w*s1v.w;
  float pd = h0.x*d0.x + h0.y*d0.y + h0.z*d0.z + h0.w*d0.w
           + h1v.x*d1v.x + h1v.y*d1v.y + h1v.z*d1v.z + h1v.w*d1v.w;
  for (int m = 1; m < 8; m <<= 1) {
    ps += __shfl_xor(ps, m);
    pd += __shfl_xor(pd, m);
  }
  if ((lane & 7) == 0) {
    const int o = wave * 4 + head;
    asrc[o] = ps;
    adst[o] = pd;
    float es = lrelu(ps + pd);
    eself[o] = es;
    emax[o]  = es;
  }
}

// Layer 2 attention prep: warp per node, 1 head x 32 channels.
__global__ void prep2(const float* __restrict__ h2,
                      const float* __restrict__ att_src,
                      const float* __restrict__ att_dst,
                      float* __restrict__ asrc, float* __restrict__ adst,
                      float* __restrict__ eself, float* __restrict__ emax,
                      int N) {
  const int wave = (blockIdx.x * blockDim.x + threadIdx.x) >> 5;
  const int lane = threadIdx.x & 31;
  if (wave >= N) return;
  float hv = h2[(size_t)wave * 32 + lane];
  float ps = hv * att_src[lane];
  float pd = hv * att_dst[lane];
  for (int m = 1; m < 32; m <<= 1) {
    ps += __shfl_xor(ps, m);
    pd += __shfl_xor(pd, m);
  }
  if (lane == 0) {
    asrc[wave] = ps;
    adst[wave] = pd;
    float es = lrelu(ps + pd);
    eself[wave] = es;
    emax[wave]  = es;
  }
}

// Edge pass A: segment max over destination (thread per edge, H heads).
template <int H>
__global__ void edge_max(const int* __restrict__ ei, int E,
                         const float* __restrict__ asrc,
                         const float* __restrict__ adst,
                         float* __restrict__ emax) {
  for (int e = blockIdx.x * blockDim.x + threadIdx.x; e < E;
       e += gridDim.x * blockDim.x) {
    const int s = ei[e], d = ei[E + e];
#pragma unroll
    for (int h = 0; h < H; ++h) {
      float v = lrelu(asrc[s * H + h] + adst[d * H + h]);
      atomicMaxF(&emax[d * H + h], v);
    }
  }
}

// Edge pass B: segment sum of exp(e - emax[d]).
template <int H>
__global__ void edge_expsum(const int* __restrict__ ei, int E,
                            const float* __restrict__ asrc,
                            const float* __restrict__ adst,
                            const float* __restrict__ emax,
                            float* __restrict__ esum) {
  for (int e = blockIdx.x * blockDim.x + threadIdx.x; e < E;
       e += gridDim.x * blockDim.x) {
    const int s = ei[e], d = ei[E + e];
#pragma unroll
    for (int h = 0; h < H; ++h) {
      const int o = d * H + h;
      float v = lrelu(asrc[s * H + h] + adst[o]);
      atomicAdd(&esum[o], __expf(v - emax[o]));
    }
  }
}

// Per (node,head): finish denominator (edges + self loop), store self weight
// and reciprocal denominator.
__global__ void denom_k(const float* __restrict__ eself,
                        const float* __restrict__ emax,
                        const float* __restrict__ esum,
                        float* __restrict__ selfw,
                        float* __restrict__ rden, int total) {
  for (int i = blockIdx.x * blockDim.x + threadIdx.x; i < total;
       i += gridDim.x * blockDim.x) {
    float ex  = __expf(eself[i] - emax[i]);
    float den = esum[i] + ex;
    float r   = 1.f / den;
    selfw[i]  = ex * r;
    rden[i]   = r;
  }
}

// Layer-1 aggregation: warp per edge; lane owns 8 contiguous channels
// (head = lane>>3). Scatter-add into the destination row.
__global__ void aggregate1(const int* __restrict__ ei, int E,
                           const float* __restrict__ asrc,
                           const float* __restrict__ adst,
                           const float* __restrict__ emax,
                           const float* __restrict__ rden,
                           const float* __restrict__ h1,
                           float* __restrict__ out1) {
  const int lane  = threadIdx.x & 31;
  const int warps = (gridDim.x * blockDim.x) >> 5;
  const int head  = lane >> 3;
  for (int e = (blockIdx.x * blockDim.x + threadIdx.x) >> 5; e < E; e += warps) {
    const int s = ei[e], d = ei[E + e];
    const int o = d * 4 + head;
    float w = __expf(lrelu(asrc[s * 4 + head] + adst[o]) - emax[o]) * rden[o];
    const float4* hv = (const float4*)(h1 + (size_t)s * 256 + lane * 8);
    float4 v0 = hv[0], v1 = hv[1];
    float* op = out1 + (size_t)d * 256 + lane * 8;
    atomicAdd(op + 0, w * v0.x); atomicAdd(op + 1, w * v0.y);
    atomicAdd(op + 2, w * v0.z); atomicAdd(op + 3, w * v0.w);
    atomicAdd(op + 4, w * v1.x); atomicAdd(op + 5, w * v1.y);
    atomicAdd(op + 6, w * v1.z); atomicAdd(op + 7, w * v1.w);
  }
}

// Layer-2 aggregation: warp per edge, lane owns 1 of 32 channels.
__global__ void aggregate2(const int* __restrict__ ei, int E,
                           const float* __restrict__ asrc,
                           const float* __restrict__ adst,
                           const float* __restrict__ emax,
                           const float* __restrict__ rden,
                           const float* __restrict__ h2,
                           float* __restrict__ out2) {
  const int lane  = threadIdx.x & 31;
  const int warps = (gridDim.x * blockDim.x) >> 5;
  for (int e = (blockIdx.x * blockDim.x + threadIdx.x) >> 5; e < E; e += warps) {
    const int s = ei[e], d = ei[E + e];
    float w = __expf(lrelu(asrc[s] + adst[d]) - emax[d]) * rden[d];
    atomicAdd(&out2[(size_t)d * 32 + lane], w * h2[(size_t)s * 32 + lane]);
  }
}

// Layer-1 epilogue: add self-loop message + bias, apply ELU.
__global__ void epilogue1(const float* __restrict__ out1,
                          const float* __restrict__ h1,
                          const float* __restrict__ selfw,
                          const float* __restrict__ b1,
                          float* __restrict__ hpost, int total) {
  for (int i = blockIdx.x * blockDim.x + threadIdx.x; i < total;
       i += gridDim.x * blockDim.x) {
    const int c = i & 255;
    const int n = i >> 8;
    const int h = c >> 6;
    float v = out1[i] + selfw[n * 4 + h] * h1[i] + b1[c];
    hpost[i] = v > 0.f ? v : (__expf(v) - 1.f);  // ELU
  }
}

// Final: self-loop message + bias (1 head, concat=False -> identity mean).
__global__ void final_k(const float* __restrict__ out2,
                        const float* __restrict__ h2,
                        const float* __restrict__ selfw,
                        const float* __restrict__ b2,
                        float* __restrict__ out, int total) {
  for (int i = blockIdx.x * blockDim.x + threadIdx.x; i < total;
       i += gridDim.x * blockDim.x) {
    const int c = i & 31;
    const int n = i >> 5;
    out[i] = out2[i] + selfw[n] * h2[i] + b2[c];
  }
}

// ---------------------------------------------------------------------------
extern "C" void kernel_launch(void* const* d_in, const int* in_sizes, int n_in,
                              void* d_out, int out_size, void* d_ws, size_t ws_size,
                              hipStream_t stream) {
  const float* x        = (const float*)d_in[0];
  const int*   ei       = (const int*)d_in[1];
  const float* W1       = (const float*)d_in[2];
  const float* att_src1 = (const float*)d_in[3];
  const float* att_dst1 = (const float*)d_in[4];
  const float* b1       = (const float*)d_in[5];
  const float* W2       = (const float*)d_in[6];
  const float* att_src2 = (const float*)d_in[7];
  const float* att_dst2 = (const float*)d_in[8];
  const float* b2       = (const float*)d_in[9];
  float* out = (float*)d_out;

  const int N = in_sizes[0] / 128;  // 50000
  const int E = in_sizes[1] / 2;    // 800000

  // Workspace layout (floats).
  float* ws     = (float*)d_ws;
  float* h1     = ws;                      // N*256
  float* hpost  = h1 + (size_t)N * 256;    // N*256
  float* out1   = hpost + (size_t)N * 256; // N*256
  float* h2     = out1 + (size_t)N * 256;  // N*32
  float* out2   = h2 + (size_t)N * 32;     // N*32
  float* p      = out2 + (size_t)N * 32;
  float* asrc1  = p; p += (size_t)N * 4;
  float* adst1  = p; p += (size_t)N * 4;
  float* eself1 = p; p += (size_t)N * 4;
  float* emax1  = p; p += (size_t)N * 4;
  float* esum1  = p; p += (size_t)N * 4;
  float* selfw1 = p; p += (size_t)N * 4;
  float* rden1  = p; p += (size_t)N * 4;
  float* asrc2  = p; p += N;
  float* adst2  = p; p += N;
  float* eself2 = p; p += N;
  float* emax2  = p; p += N;
  float* esum2  = p; p += N;
  float* selfw2 = p; p += N;
  float* rden2  = p; p += N;

  const int BT = 256;  // 8 waves per block (wave32)

  // ---- Layer 1 ----
  {
    // 16x64 strip per wave: (N/16)*(256/64) waves.
    int tiles = (N / 16) * (256 / 64);
    int blocks = (tiles * 32 + BT - 1) / BT;
    gemm_f32_wmma<4><<<blocks, BT, 0, stream>>>(x, W1, h1, N, 128, 256);
  }
  hipMemsetAsync(out1, 0, (size_t)N * 256 * sizeof(float), stream);
  hipMemsetAsync(esum1, 0, (size_t)N * 4 * sizeof(float), stream);

  prep1<<<(N * 32 + BT - 1) / BT, BT, 0, stream>>>(h1, att_src1, att_dst1,
                                                   asrc1, adst1, eself1, emax1, N);
  {
    int blocks = (E + BT - 1) / BT;
    edge_max<4><<<blocks, BT, 0, stream>>>(ei, E, asrc1, adst1, emax1);
    edge_expsum<4><<<blocks, BT, 0, stream>>>(ei, E, asrc1, adst1, emax1, esum1);
  }
  denom_k<<<(N * 4 + BT - 1) / BT, BT, 0, stream>>>(eself1, emax1, esum1,
                                                    selfw1, rden1, N * 4);
  aggregate1<<<(E * 32 + BT - 1) / BT, BT, 0, stream>>>(ei, E, asrc1, adst1,
                                                        emax1, rden1, h1, out1);
  epilogue1<<<(N * 256 + BT - 1) / BT, BT, 0, stream>>>(out1, h1, selfw1, b1,
                                                        hpost, N * 256);

  // ---- Layer 2 ----
  {
    // 16x32 strip per wave: (N/16)*(32/32) waves.
    int tiles = (N / 16) * (32 / 32);
    int blocks = (tiles * 32 + BT - 1) / BT;
    gemm_f32_wmma<2><<<blocks, BT, 0, stream>>>(hpost, W2, h2, N, 256, 32);
  }
  hipMemsetAsync(out2, 0, (size_t)N * 32 * sizeof(float), stream);
  hipMemsetAsync(esum2, 0, (size_t)N * sizeof(float), stream);

  prep2<<<(N * 32 + BT - 1) / BT, BT, 0, stream>>>(h2, att_src2, att_dst2,
                                                   asrc2, adst2, eself2, emax2, N);
  {
    int blocks = (E + BT - 1) / BT;
    edge_max<1><<<blocks, BT, 0, stream>>>(ei, E, asrc2, adst2, emax2);
    edge_expsum<1><<<blocks, BT, 0, stream>>>(ei, E, asrc2, adst2, emax2, esum2);
  }
  denom_k<<<(N + BT - 1) / BT, BT, 0, stream>>>(eself2, emax2, esum2,
                                                selfw2, rden2, N);
  aggregate2<<<(E * 32 + BT - 1) / BT, BT, 0, stream>>>(ei, E, asrc2, adst2,
                                                        emax2, rden2, h2, out2);
  final_k<<<(N * 32 + BT - 1) / BT, BT, 0, stream>>>(out2, h2, selfw2, b2,
                                                     out, N * 32);
}